// StiglerPDHGSolver_47433618817421
// MI455X (gfx1250) — compile-verified
//
#include <hip/hip_runtime.h>

#define GRID    128
#define BLOCK   256
#define M_DIM   2048
#define N_DIM   8192
#define NITERS  1000
#define TAU_F   1e-4f
#define SIGMA_F 1e-4f

typedef __attribute__((ext_vector_type(2))) float v2f;
typedef __attribute__((ext_vector_type(8))) float v8f;

// Monotonic grid barrier: counter only ever increments; target = bar_no * GRID.
__device__ __forceinline__ void grid_barrier(unsigned* cnt, unsigned target) {
    __threadfence();          // make this WG's global stores visible device-wide
    __syncthreads();
    if (threadIdx.x == 0) {
        __hip_atomic_fetch_add(cnt, 1u, __ATOMIC_RELEASE, __HIP_MEMORY_SCOPE_AGENT);
        while (__hip_atomic_load(cnt, __ATOMIC_ACQUIRE, __HIP_MEMORY_SCOPE_AGENT) < target) {
            __builtin_amdgcn_s_sleep(1);
        }
    }
    __syncthreads();
    __threadfence();          // discard stale cached lines before reading peers' data
}

__global__ __launch_bounds__(BLOCK) void pdhg_persistent(
    const float* __restrict__ cvec, const float* __restrict__ A,
    const float* __restrict__ bvec, float* __restrict__ out,
    float* __restrict__ ws)
{
    float*    x    = ws;                              // [N_DIM]
    float*    xbar = ws + N_DIM;                      // [N_DIM]
    float*    y    = ws + 2 * N_DIM;                  // [M_DIM]
    float*    lam  = ws + 2 * N_DIM + M_DIM;          // [M_DIM]
    unsigned* cnt  = (unsigned*)(ws + 2 * N_DIM + 2 * M_DIM);

    const int tid  = threadIdx.x;
    const int lane = tid & 31;
    const int wave = tid >> 5;                        // 0..7
    const int wg   = blockIdx.x;

    // ---- phase-1 geometry: 16 rows per WG, K split across 8 waves (1024 each)
    const int r0   = wg * 16;
    const int xoff = 2 * (lane >> 4);                 // lanes 16-31 cover K+2,K+3
    const float* Arow = A + (size_t)(r0 + (lane & 15)) * N_DIM + xoff;
    const int k0 = wave * (N_DIM / (BLOCK / 32));     // wave * 1024

    // ---- phase-2 geometry: 64 cols per WG as 32 column pairs, 8 m-groups x 256
    const int pair   = tid & 31;
    const int col0   = wg * 64 + pair * 2;
    const int mstart = (tid >> 5) * (M_DIM / 8);      // 256 rows per group
    const float* Ac  = A + col0;

    __shared__ float xs[N_DIM];                       // staged x_bar (32 KB)
    __shared__ float ys[M_DIM];                       // staged y / lam (8 KB)
    __shared__ float redp[(BLOCK / 32) * 16];         // per-wave row partials
    __shared__ v2f   red2[BLOCK];                     // phase-2 reduction (2 KB)

    unsigned bar = 0;
    const v2f zero2 = {0.f, 0.f};

    for (int it = 0; it < NITERS; ++it) {
        // ============ phase 1: y = min(0, y + sigma*(A @ xbar) - sigma*b) ======
        // stage x_bar into LDS once; B-operand then comes from ds broadcasts,
        // keeping the whole VMEM path for the A stream.
        for (int i = tid * 2; i < N_DIM; i += 2 * BLOCK)
            *(v2f*)(xs + i) = *(const v2f*)(xbar + i);
        __syncthreads();

        v8f acc0 = {}; v8f acc1 = {};
        #pragma unroll 4
        for (int kk = k0; kk < k0 + N_DIM / 8; kk += 8) {
            v2f a0  = *(const v2f*)(Arow + kk);
            v2f a1  = *(const v2f*)(Arow + kk + 4);
            v2f xb0 = *(const v2f*)(xs + kk + xoff);       // ds_load_b64
            v2f xb1 = *(const v2f*)(xs + kk + 4 + xoff);
            v2f b0 = ((lane & 15) == 0) ? xb0 : zero2;     // x in B column 0 only
            v2f b1 = ((lane & 15) == 0) ? xb1 : zero2;
            acc0 = __builtin_amdgcn_wmma_f32_16x16x4_f32(
                       false, a0, false, b0, (short)0, acc0, false, false);
            acc1 = __builtin_amdgcn_wmma_f32_16x16x4_f32(
                       false, a1, false, b1, (short)0, acc1, false, false);
        }
        // D col 0: lane 0 holds rows r0..r0+7 (vgpr 0..7), lane 16 rows r0+8..r0+15
        if (lane == 0) {
            #pragma unroll
            for (int i = 0; i < 8; ++i) redp[wave * 16 + i] = acc0[i] + acc1[i];
        } else if (lane == 16) {
            #pragma unroll
            for (int i = 0; i < 8; ++i) redp[wave * 16 + 8 + i] = acc0[i] + acc1[i];
        }
        __syncthreads();
        if (tid < 16) {
            float s = 0.f;
            #pragma unroll
            for (int v = 0; v < BLOCK / 32; ++v) s += redp[v * 16 + tid];
            const int row = r0 + tid;
            y[row] = fminf(0.f, y[row] + SIGMA_F * s - SIGMA_F * bvec[row]);
        }
        grid_barrier(cnt, (++bar) * GRID);

        // ============ phase 2: x = relu(x - tau*(A^T @ y) - tau*c) =============
        for (int i = tid; i < M_DIM; i += BLOCK) ys[i] = y[i];
        __syncthreads();
        v2f acc = zero2;
        #pragma unroll 8
        for (int m = mstart; m < mstart + M_DIM / 8; ++m) {
            v2f av   = *(const v2f*)(Ac + (size_t)m * N_DIM);   // 256B per wave
            float yv = ys[m];
            acc.x = fmaf(av.x, yv, acc.x);
            acc.y = fmaf(av.y, yv, acc.y);
        }
        red2[tid] = acc;
        __syncthreads();
        if (tid < 32) {
            v2f t = zero2;
            #pragma unroll
            for (int g = 0; g < 8; ++g) {
                t.x += red2[tid + 32 * g].x;
                t.y += red2[tid + 32 * g].y;
            }
            const int n0 = wg * 64 + tid * 2;
            float xo0 = x[n0], xo1 = x[n0 + 1];
            float xn0 = fmaxf(0.f, xo0 - TAU_F * t.x - TAU_F * cvec[n0]);
            float xn1 = fmaxf(0.f, xo1 - TAU_F * t.y - TAU_F * cvec[n0 + 1]);
            x[n0]     = xn0;          x[n0 + 1]    = xn1;
            xbar[n0]  = 2.f * xn0 - xo0;
            xbar[n0 + 1] = 2.f * xn1 - xo1;      // theta = 1
        }
        grid_barrier(cnt, (++bar) * GRID);
    }

    // ============ tail: lam = relu(-y); nu = relu(c - A^T @ lam) ===============
    if (tid < 16) {
        const int row = r0 + tid;
        float l = fmaxf(0.f, -y[row]);
        lam[row] = l;
        out[N_DIM + row] = l;                          // lam output
    }
    if (tid < 32) {
        const int n0 = wg * 64 + tid * 2;
        out[n0]     = x[n0];                           // x output
        out[n0 + 1] = x[n0 + 1];
    }
    grid_barrier(cnt, (++bar) * GRID);

    for (int i = tid; i < M_DIM; i += BLOCK) ys[i] = lam[i];
    __syncthreads();
    v2f acc = zero2;
    #pragma unroll 8
    for (int m = mstart; m < mstart + M_DIM / 8; ++m) {
        v2f av   = *(const v2f*)(Ac + (size_t)m * N_DIM);
        float yv = ys[m];
        acc.x = fmaf(av.x, yv, acc.x);
        acc.y = fmaf(av.y, yv, acc.y);
    }
    red2[tid] = acc;
    __syncthreads();
    if (tid < 32) {
        v2f t = zero2;
        #pragma unroll
        for (int g = 0; g < 8; ++g) {
            t.x += red2[tid + 32 * g].x;
            t.y += red2[tid + 32 * g].y;
        }
        const int n0 = wg * 64 + tid * 2;
        out[N_DIM + M_DIM + n0]     = fmaxf(0.f, cvec[n0]     - t.x);  // nu
        out[N_DIM + M_DIM + n0 + 1] = fmaxf(0.f, cvec[n0 + 1] - t.y);
    }
}

extern "C" void kernel_launch(void* const* d_in, const int* in_sizes, int n_in,
                              void* d_out, int out_size, void* d_ws, size_t ws_size,
                              hipStream_t stream) {
    const float* c = (const float*)d_in[0];
    const float* A = (const float*)d_in[1];
    const float* b = (const float*)d_in[2];
    float* out = (float*)d_out;
    float* ws  = (float*)d_ws;

    // zero x, xbar, y, lam and the barrier counter (capture-safe memset node)
    const size_t init_bytes = (size_t)(2 * N_DIM + 2 * M_DIM + 1) * sizeof(float);
    hipMemsetAsync(d_ws, 0, init_bytes, stream);

    pdhg_persistent<<<GRID, BLOCK, 0, stream>>>(c, A, b, out, ws);
}